// IntraDaySnapshotEncoder_21534966022952
// MI455X (gfx1250) — compile-verified
//
#include <hip/hip_runtime.h>

// ---------------------------------------------------------------------------
// RGAT forward for MI455X (gfx1250, wave32).
// All dense linears run on the f32 matrix pipe via V_WMMA_F32_16X16X4_F32.
// B (weight) panels are staged per-block into LDS (pair-interleaved) so each
// lane fetches its B fragment with one ds_load_b64; A rows are clamped instead
// of predicated so the WMMA K-loop has zero control flow.
// Edge phase uses hoisted per-node linears + hw float atomics for
// segment-softmax / scatter aggregation.
// ---------------------------------------------------------------------------

#define NS 20000
#define NB 500
#define NI 100
#define E_SS 320000
#define E_SB 100000
#define E_BS 100000
#define E_SI 60000
#define E_IS 60000
#define E_II 4000

typedef float v8f __attribute__((ext_vector_type(8)));
typedef float v2f __attribute__((ext_vector_type(2)));

// ---- ordered-uint encoding so atomicMax(u32) implements float max ---------
__device__ __forceinline__ unsigned enc_ord(float f) {
  unsigned u = __float_as_uint(f);
  return (u & 0x80000000u) ? ~u : (u | 0x80000000u);
}
__device__ __forceinline__ float dec_ord(unsigned e) {
  unsigned u = (e & 0x80000000u) ? (e & 0x7FFFFFFFu) : ~e;
  return __uint_as_float(u);
}

// Stage W[0:K, n0:n0+16] into LDS, pair-interleaved:
// lds[(k>>1)*32 + c*2 + (k&1)] = W[k][n0+c]  ->  B frag = one b64 per lane.
__device__ __forceinline__ void stage_B(float* ldsB, const float* __restrict__ W,
                                        int K, int n0) {
  for (int i = threadIdx.x; i < K * 16; i += 256) {
    int k = i >> 4, c = i & 15;
    ldsB[(k >> 1) * 32 + c * 2 + (k & 1)] = W[(size_t)k * 128 + n0 + c];
  }
  __syncthreads();
}

// ---------------------------------------------------------------------------
// Generic GEMM: C[M,128] = act(A[M,K] @ W[K,128] + bias), K % 4 == 0, K <= 256.
// One wave per 16x16 output tile, 8 waves / 256-thread block (same n0).
// A frag (16x4 f32): lanes 0-15 hold K=k,k+1 ; lanes 16-31 hold K=k+2,k+3.
// C frag (v8f): vgpr i -> row m0+i (lanes 0-15) / m0+8+i (lanes 16-31).
// Rows >= M are clamped (their D rows are never stored).
// act: 0 = none, 1 = relu, 2 = tanh
// ---------------------------------------------------------------------------
__global__ __launch_bounds__(256) void gemm128(const float* __restrict__ A,
    const float* __restrict__ W, const float* __restrict__ bias,
    float* __restrict__ C, int M, int K, int act)
{
  __shared__ float ldsB[4096];
  const int n0 = blockIdx.y * 16;
  stage_B(ldsB, W, K, n0);

  const int lane = threadIdx.x & 31;
  const int wv   = threadIdx.x >> 5;
  const int m0   = (blockIdx.x * 8 + wv) * 16;
  if (m0 >= M) return;                     // wave-uniform, after the barrier
  const int row = lane & 15;
  const int kh  = lane >> 4;               // 0 or 1
  const int rc  = min(m0 + row, M - 1);    // clamped: garbage rows unused
  const float* Ar = A + (size_t)rc * K;
  const float* Bb = ldsB + row * 2 + kh * 32;
  v8f acc = {0.f,0.f,0.f,0.f,0.f,0.f,0.f,0.f};
  for (int k = 0; k < K; k += 4) {
    v2f a = *(const v2f*)(Ar + k + 2 * kh);
    v2f b = *(const v2f*)(Bb + (k >> 1) * 32);
    acc = __builtin_amdgcn_wmma_f32_16x16x4_f32(false, a, false, b,
                                                (short)0, acc, false, false);
  }
  const int col = n0 + row;
  const float bv = bias ? bias[col] : 0.f;
#pragma unroll
  for (int i = 0; i < 8; ++i) {
    int rr = m0 + i + 8 * kh;
    if (rr < M) {
      float v = acc[i] + bv;
      if (act == 1)      v = v > 0.f ? v : 0.f;
      else if (act == 2) v = tanhf(v);
      C[(size_t)rr * 128 + col] = v;
    }
  }
}

// C[M,128] = A_cat[M,256] @ W[256,128] + bias, with A_cat = [A1 | A2]
__global__ __launch_bounds__(256) void gemm128_cat(const float* __restrict__ A1,
    const float* __restrict__ A2, const float* __restrict__ W,
    const float* __restrict__ bias, float* __restrict__ C, int M)
{
  __shared__ float ldsB[4096];
  const int n0 = blockIdx.y * 16;
  stage_B(ldsB, W, 256, n0);

  const int lane = threadIdx.x & 31;
  const int wv   = threadIdx.x >> 5;
  const int m0   = (blockIdx.x * 8 + wv) * 16;
  if (m0 >= M) return;
  const int row = lane & 15;
  const int kh  = lane >> 4;
  const int rc  = min(m0 + row, M - 1);
  const float* A1r = A1 + (size_t)rc * 128;
  const float* A2r = A2 + (size_t)rc * 128;
  const float* Bb  = ldsB + row * 2 + kh * 32;
  v8f acc = {0.f,0.f,0.f,0.f,0.f,0.f,0.f,0.f};
  for (int k = 0; k < 256; k += 4) {
    // k is a multiple of 4, so ka = k + 2*kh stays within one half: uniform sel
    const float* Ar = (k < 128) ? A1r : A2r;
    v2f a = *(const v2f*)(Ar + ((k & 127) + 2 * kh));
    v2f b = *(const v2f*)(Bb + (k >> 1) * 32);
    acc = __builtin_amdgcn_wmma_f32_16x16x4_f32(false, a, false, b,
                                                (short)0, acc, false, false);
  }
  const int col = n0 + row;
  const float bv = bias ? bias[col] : 0.f;
#pragma unroll
  for (int i = 0; i < 8; ++i) {
    int rr = m0 + i + 8 * kh;
    if (rr < M) C[(size_t)rr * 128 + col] = acc[i] + bv;
  }
}

// Meta "stacked" GEMM: logical A row r = path[r&3][r>>2], K = 128, act = tanh
__global__ __launch_bounds__(256) void gemm128_p4(const float* __restrict__ A0,
    const float* __restrict__ A1, const float* __restrict__ A2,
    const float* __restrict__ A3, const float* __restrict__ W,
    const float* __restrict__ bias, float* __restrict__ C, int M)
{
  __shared__ float ldsB[4096];
  const int n0 = blockIdx.y * 16;
  stage_B(ldsB, W, 128, n0);

  const int lane = threadIdx.x & 31;
  const int wv   = threadIdx.x >> 5;
  const int m0   = (blockIdx.x * 8 + wv) * 16;
  if (m0 >= M) return;
  const int row = lane & 15;
  const int kh  = lane >> 4;
  const int rc  = min(m0 + row, M - 1);
  const int p   = rc & 3;
  const float* Ap = (p == 0) ? A0 : (p == 1) ? A1 : (p == 2) ? A2 : A3;
  const float* Ar = Ap + (size_t)(rc >> 2) * 128;
  const float* Bb = ldsB + row * 2 + kh * 32;
  v8f acc = {0.f,0.f,0.f,0.f,0.f,0.f,0.f,0.f};
  for (int k = 0; k < 128; k += 4) {
    v2f a = *(const v2f*)(Ar + k + 2 * kh);
    v2f b = *(const v2f*)(Bb + (k >> 1) * 32);
    acc = __builtin_amdgcn_wmma_f32_16x16x4_f32(false, a, false, b,
                                                (short)0, acc, false, false);
  }
  const int col = n0 + row;
  const float bv = bias[col];
#pragma unroll
  for (int i = 0; i < 8; ++i) {
    int rr = m0 + i + 8 * kh;
    if (rr < M) C[(size_t)rr * 128 + col] = tanhf(acc[i] + bv);
  }
}

// ---- small helper kernels -------------------------------------------------

// WB[128,128] = blockdiag(Wm[32,32]) over 4 heads
__global__ void build_wm_block(const float* __restrict__ Wm, float* __restrict__ WB) {
  int i = blockIdx.x * blockDim.x + threadIdx.x;
  if (i >= 128 * 128) return;
  int r = i >> 7, c = i & 127;
  WB[i] = ((r >> 5) == (c >> 5)) ? Wm[(r & 31) * 32 + (c & 31)] : 0.f;
}

// beM[c] = sum_i be[h*32+i]*Wm[i][c%32] + bm[c%32]   (h = c/32)
__global__ void build_beM(const float* __restrict__ be, const float* __restrict__ Wm,
                          const float* __restrict__ bm, float* __restrict__ out) {
  int c = threadIdx.x;
  int h = c >> 5, j = c & 31;
  float s = bm[j];
  for (int i = 0; i < 32; ++i) s += be[h * 32 + i] * Wm[i * 32 + j];
  out[c] = s;
}

// per-node partial attention score: out[n,h] = sum_k tanh(F[n,h*32+k]) * Wa[k]
__global__ void node_score(const float* __restrict__ F, const float* __restrict__ Wa,
                           float* __restrict__ out, int N) {
  int t = blockIdx.x * blockDim.x + threadIdx.x;
  if (t >= N * 4) return;
  int n = t >> 2, h = t & 3;
  const float* f = F + (size_t)n * 128 + h * 32;
  float s = 0.f;
#pragma unroll
  for (int k = 0; k < 32; ++k) s += tanhf(f[k]) * Wa[k];
  out[t] = s;
}

__global__ void fill_u32(unsigned* __restrict__ p, unsigned v, int n) {
  int t = blockIdx.x * blockDim.x + threadIdx.x;
  if (t < n) p[t] = v;
}

// per-(edge,head) raw score + atomic segment max over dst
__global__ void edge_score(const int* __restrict__ srcI, const int* __restrict__ dstI,
                           const float* __restrict__ ea, int ed,
                           const float* __restrict__ We, const float* __restrict__ be,
                           const float* __restrict__ Wa64,
                           const float* __restrict__ sA, const float* __restrict__ sD,
                           const float* __restrict__ ba,
                           float* __restrict__ scores, unsigned* __restrict__ mx, int E) {
  int t = blockIdx.x * blockDim.x + threadIdx.x;
  if (t >= E * 4) return;
  int e = t >> 2, h = t & 3;
  float av[8];
  for (int d = 0; d < ed; ++d) av[d] = ea[(size_t)e * ed + d];
  float s = 0.f;
  for (int k = 0; k < 32; ++k) {
    float v = be[h * 32 + k];
    for (int d = 0; d < ed; ++d) v += av[d] * We[(size_t)d * 128 + h * 32 + k];
    s += tanhf(v) * Wa64[k];
  }
  s += sA[(size_t)srcI[e] * 4 + h] + sD[(size_t)dstI[e] * 4 + h] + ba[0];
  scores[t] = s;
  atomicMax(&mx[(size_t)dstI[e] * 4 + h], enc_ord(s));
}

// exp(score - segmax) in place + atomic segment sum
__global__ void edge_exp(const int* __restrict__ dstI, const unsigned* __restrict__ mx,
                         float* __restrict__ scores, float* __restrict__ sm, int E) {
  int t = blockIdx.x * blockDim.x + threadIdx.x;
  if (t >= E * 4) return;
  int e = t >> 2, h = t & 3;
  float m = dec_ord(mx[(size_t)dstI[e] * 4 + h]);
  float v = __expf(scores[t] - m);
  scores[t] = v;
  unsafeAtomicAdd(&sm[(size_t)dstI[e] * 4 + h], v);
}

// one wave per edge: agg[dst] += alpha * (nodeM[src] + ea@We_m + beM)
__global__ __launch_bounds__(256) void edge_agg(const int* __restrict__ srcI,
    const int* __restrict__ dstI, const float* __restrict__ ea, int ed,
    const float* __restrict__ WeM, const float* __restrict__ beMv,
    const float* __restrict__ nodeM, const float* __restrict__ ex,
    const float* __restrict__ sm, float* __restrict__ agg, int E) {
  int w = (blockIdx.x * blockDim.x + threadIdx.x) >> 5;
  int lane = threadIdx.x & 31;
  if (w >= E) return;
  int s = srcI[w], d = dstI[w];
  float av[8];
  for (int dd = 0; dd < ed; ++dd) av[dd] = ea[(size_t)w * ed + dd];
#pragma unroll
  for (int h = 0; h < 4; ++h) {
    int c = h * 32 + lane;
    float v = beMv[c] + nodeM[(size_t)s * 128 + c];
    for (int dd = 0; dd < ed; ++dd) v += av[dd] * WeM[(size_t)dd * 128 + c];
    float alpha = ex[(size_t)w * 4 + h] / sm[(size_t)d * 4 + h];
    unsafeAtomicAdd(&agg[(size_t)d * 128 + c], alpha * v);
  }
}

// mean over 4 heads: out[n,k] = mean_h agg[n,h*32+k]
__global__ void head_mean(const float* __restrict__ agg, float* __restrict__ out, int N) {
  int t = blockIdx.x * blockDim.x + threadIdx.x;
  if (t >= N * 32) return;
  int n = t >> 5, k = t & 31;
  const float* a = agg + (size_t)n * 128;
  out[t] = 0.25f * (a[k] + a[32 + k] + a[64 + k] + a[96 + k]);
}

// out = LN((addBase ? base : 0) + upd) * g + b   (one 128-thread block per row)
__global__ void res_ln(const float* __restrict__ base, const float* __restrict__ upd,
                       const float* __restrict__ g, const float* __restrict__ b,
                       float* __restrict__ out, int addBase) {
  __shared__ float red[128];
  int n = blockIdx.x, t = threadIdx.x;
  float x = upd[(size_t)n * 128 + t];
  if (addBase) x += base[(size_t)n * 128 + t];
  red[t] = x; __syncthreads();
  for (int s = 64; s > 0; s >>= 1) { if (t < s) red[t] += red[t + s]; __syncthreads(); }
  float mean = red[0] * (1.f / 128.f);
  __syncthreads();
  float dx = x - mean;
  red[t] = dx * dx; __syncthreads();
  for (int s = 64; s > 0; s >>= 1) { if (t < s) red[t] += red[t + s]; __syncthreads(); }
  float var = red[0] * (1.f / 128.f);
  out[(size_t)n * 128 + t] = dx * rsqrtf(var + 1e-5f) * g[t] + b[t];
}

// meta-path attention: scores from hbuf (= tanh(stacked@Wp+bp)) dot ws,
// softmax over the 4 paths, weighted combine of raw paths.
__global__ void meta_combine(const float* __restrict__ hbuf, const float* __restrict__ wsv,
                             const float* __restrict__ p0, const float* __restrict__ p1,
                             const float* __restrict__ p2, const float* __restrict__ p3,
                             float* __restrict__ out, int N) {
  __shared__ float red[128];
  __shared__ float sc[4];
  int n = blockIdx.x, t = threadIdx.x;
  for (int p = 0; p < 4; ++p) {
    red[t] = hbuf[((size_t)n * 4 + p) * 128 + t] * wsv[t];
    __syncthreads();
    for (int s = 64; s > 0; s >>= 1) { if (t < s) red[t] += red[t + s]; __syncthreads(); }
    if (t == 0) sc[p] = red[0];
    __syncthreads();
  }
  float m = fmaxf(fmaxf(sc[0], sc[1]), fmaxf(sc[2], sc[3]));
  float e0 = __expf(sc[0] - m), e1 = __expf(sc[1] - m);
  float e2 = __expf(sc[2] - m), e3 = __expf(sc[3] - m);
  float inv = 1.f / (e0 + e1 + e2 + e3);
  size_t i = (size_t)n * 128 + t;
  out[i] = (e0 * p0[i] + e1 * p1[i] + e2 * p2[i] + e3 * p3[i]) * inv;
}

// ---------------------------------------------------------------------------
// Host side
// ---------------------------------------------------------------------------

struct RgaP {
  const float *Wa, *Wdst, *We, *Wm, *Wmg, *Wo, *Wsrc;
  const float *ba, *bdst, *be, *bln, *bm, *bmg, *bo, *bsrc, *g;
};

// Param pytree assumed flattened with JAX sorted-key order after the 15
// positional tensors.  rga leaf order (sorted):
// Wa,Wdst,We,Wm,Wmg,Wo,Wsrc,ba,bdst,be,bln,bm,bmg,bo,bsrc,g
static RgaP get_rga(void* const* d_in, int base) {
  RgaP p;
  p.Wa   = (const float*)d_in[base + 0];
  p.Wdst = (const float*)d_in[base + 1];
  p.We   = (const float*)d_in[base + 2];
  p.Wm   = (const float*)d_in[base + 3];
  p.Wmg  = (const float*)d_in[base + 4];
  p.Wo   = (const float*)d_in[base + 5];
  p.Wsrc = (const float*)d_in[base + 6];
  p.ba   = (const float*)d_in[base + 7];
  p.bdst = (const float*)d_in[base + 8];
  p.be   = (const float*)d_in[base + 9];
  p.bln  = (const float*)d_in[base + 10];
  p.bm   = (const float*)d_in[base + 11];
  p.bmg  = (const float*)d_in[base + 12];
  p.bo   = (const float*)d_in[base + 13];
  p.bsrc = (const float*)d_in[base + 14];
  p.g    = (const float*)d_in[base + 15];
  return p;
}

extern "C" void kernel_launch(void* const* d_in, const int* in_sizes, int n_in,
                              void* d_out, int out_size, void* d_ws, size_t ws_size,
                              hipStream_t stream) {
  (void)in_sizes; (void)n_in; (void)out_size; (void)ws_size;

  // --- inputs (setup_inputs insertion order) ---
  const float* stockF = (const float*)d_in[0];
  const float* bankF  = (const float*)d_in[1];
  const float* indF   = (const float*)d_in[2];
  const int*   eiSS = (const int*)d_in[3];  const float* eaSS = (const float*)d_in[4];
  const int*   eiSB = (const int*)d_in[5];  const float* eaSB = (const float*)d_in[6];
  const int*   eiBS = (const int*)d_in[7];  const float* eaBS = (const float*)d_in[8];
  const int*   eiSI = (const int*)d_in[9];  const float* eaSI = (const float*)d_in[10];
  const int*   eiIS = (const int*)d_in[11]; const float* eaIS = (const float*)d_in[12];
  const int*   eiII = (const int*)d_in[13]; const float* eaII = (const float*)d_in[14];

  // --- param leaves (sorted pytree order) ---
  // 15..18 bank_enc {W1,W2,b1,b2}; 19..22 industry_enc; 23.. layers; 215 meta; 222 stock_enc
  const float* beW1 = (const float*)d_in[15]; const float* beW2 = (const float*)d_in[16];
  const float* beb1 = (const float*)d_in[17]; const float* beb2 = (const float*)d_in[18];
  const float* ieW1 = (const float*)d_in[19]; const float* ieW2 = (const float*)d_in[20];
  const float* ieb1 = (const float*)d_in[21]; const float* ieb2 = (const float*)d_in[22];
  const float* mWo  = (const float*)d_in[215]; const float* mWp  = (const float*)d_in[216];
  const float* mbln = (const float*)d_in[217]; const float* mbo  = (const float*)d_in[218];
  const float* mbp  = (const float*)d_in[219]; const float* mg   = (const float*)d_in[220];
  const float* mws  = (const float*)d_in[221];
  const float* seW1 = (const float*)d_in[222]; const float* seW2 = (const float*)d_in[223];
  const float* seb1 = (const float*)d_in[224]; const float* seb2 = (const float*)d_in[225];

  // --- workspace carve ---
  float* wsp = (float*)d_ws;
  size_t off = 0;
  auto alloc = [&](size_t n) { float* p = wsp + off; off += n; return p; };
  float* sh    = alloc((size_t)NS * 128);
  float* bhv   = alloc((size_t)NB * 128);
  float* ihv   = alloc((size_t)NI * 128);
  float* path1 = alloc((size_t)NS * 128);
  float* path2 = alloc((size_t)NS * 128);
  float* path3 = alloc((size_t)NS * 128);
  float* WmB   = alloc(128 * 128);
  float* WeM   = alloc(16 * 128);
  float* beMv  = alloc(128);
  float* nodeA = alloc((size_t)NS * 128);     // also meta hbuf head
  float* nodeD = alloc((size_t)NS * 128);
  float* nodeM = alloc((size_t)NS * 128);
  float* sA    = alloc((size_t)NS * 4);
  float* sD    = alloc((size_t)NS * 4);
  float* scores= alloc((size_t)E_SS * 4);
  float* mxv   = alloc((size_t)NS * 4);
  float* smv   = alloc((size_t)NS * 4);
  float* agg   = alloc((size_t)NS * 128);
  float* magg  = alloc((size_t)NS * 32);
  float* agglin= alloc((size_t)NS * 128);     // also meta combine out
  float* updm  = alloc((size_t)NS * 128);
  float* hbuf  = nodeA;                       // meta hbuf aliases dead RGA scratch

  auto gemm = [&](const float* A, const float* W, const float* bias, float* C,
                  int M, int K, int act) {
    dim3 g((unsigned)((M + 127) / 128), 8);
    gemm128<<<g, 256, 0, stream>>>(A, W, bias, C, M, K, act);
  };

  auto run_rga = [&](const RgaP& p, int ed, const float* srcX, int Nsrc,
                     const float* dstX, int Ndst, const int* ei, const float* ea,
                     int E, float* outBuf) {
    const int* srcI = ei;
    const int* dstI = ei + E;
    build_wm_block<<<64, 256, 0, stream>>>(p.Wm, WmB);
    gemm(p.We, WmB, nullptr, WeM, ed, 128, 0);              // We_m = We @ blockdiag(Wm)
    build_beM<<<1, 128, 0, stream>>>(p.be, p.Wm, p.bm, beMv);
    gemm(srcX, p.Wsrc, p.bsrc, nodeA, Nsrc, 128, 0);        // src_f per node
    gemm(dstX, p.Wdst, p.bdst, nodeD, Ndst, 128, 0);        // dst_f per node
    gemm(nodeA, WmB, nullptr, nodeM, Nsrc, 128, 0);         // src msg per node
    node_score<<<(Nsrc * 4 + 255) / 256, 256, 0, stream>>>(nodeA, p.Wa, sA, Nsrc);
    node_score<<<(Ndst * 4 + 255) / 256, 256, 0, stream>>>(nodeD, p.Wa + 32, sD, Ndst);
    fill_u32<<<(Ndst * 4 + 255) / 256, 256, 0, stream>>>((unsigned*)mxv, 0x007FFFFFu, Ndst * 4);
    fill_u32<<<(Ndst * 4 + 255) / 256, 256, 0, stream>>>((unsigned*)smv, 0u, Ndst * 4);
    fill_u32<<<(Ndst * 128 + 255) / 256, 256, 0, stream>>>((unsigned*)agg, 0u, Ndst * 128);
    edge_score<<<(E * 4 + 255) / 256, 256, 0, stream>>>(srcI, dstI, ea, ed, p.We, p.be,
        p.Wa + 64, sA, sD, p.ba, scores, (unsigned*)mxv, E);
    edge_exp<<<(E * 4 + 255) / 256, 256, 0, stream>>>(dstI, (unsigned*)mxv, scores, smv, E);
    edge_agg<<<(E + 7) / 8, 256, 0, stream>>>(srcI, dstI, ea, ed, WeM, beMv, nodeM,
        scores, smv, agg, E);
    head_mean<<<(Ndst * 32 + 255) / 256, 256, 0, stream>>>(agg, magg, Ndst);
    gemm(magg, p.Wmg, p.bmg, agglin, Ndst, 32, 0);
    {
      dim3 g((unsigned)((Ndst + 127) / 128), 8);
      gemm128_cat<<<g, 256, 0, stream>>>(dstX, agglin, p.Wo, p.bo, nodeD, Ndst);
    }
    res_ln<<<Ndst, 128, 0, stream>>>(dstX, nodeD, p.g, p.bln, outBuf, 1);
  };

  // --- node encoders (Linear -> ReLU -> Linear) ---
  gemm(stockF, seW1, seb1, nodeA, NS, 64, 1);
  gemm(nodeA, seW2, seb2, sh, NS, 128, 0);
  gemm(bankF, beW1, beb1, nodeA, NB, 32, 1);
  gemm(nodeA, beW2, beb2, bhv, NB, 128, 0);
  gemm(indF, ieW1, ieb1, nodeA, NI, 16, 1);
  gemm(nodeA, ieW2, ieb2, ihv, NI, 128, 0);

  // --- layers ---
  for (int L = 0; L < 2; ++L) {
    // layer dict sorted rel order: BS,II,IS,SB,SI,SS ; 16 leaves each
    const int lb = 23 + L * 96;
    RgaP pBS = get_rga(d_in, lb + 0);
    RgaP pII = get_rga(d_in, lb + 16);
    RgaP pIS = get_rga(d_in, lb + 32);
    RgaP pSB = get_rga(d_in, lb + 48);
    RgaP pSI = get_rga(d_in, lb + 64);
    RgaP pSS = get_rga(d_in, lb + 80);

    run_rga(pSS, 8, sh, NS, sh, NS, eiSS, eaSS, E_SS, path1);
    run_rga(pSB, 4, bhv, NB, sh, NS, eiSB, eaSB, E_SB, path2);
    run_rga(pSI, 4, ihv, NI, sh, NS, eiSI, eaSI, E_SI, path3);

    // meta-path attention over {sh, path1, path2, path3}
    {
      dim3 g((unsigned)((NS * 4 + 127) / 128), 8);
      gemm128_p4<<<g, 256, 0, stream>>>(sh, path1, path2, path3, mWp, mbp, hbuf, NS * 4);
    }
    meta_combine<<<NS, 128, 0, stream>>>(hbuf, mws, sh, path1, path2, path3, agglin, NS);
    gemm(agglin, mWo, mbo, updm, NS, 128, 0);
    float* shTarget = (L == 0) ? sh : (float*)d_out;   // final layer writes output
    res_ln<<<NS, 128, 0, stream>>>(nullptr, updm, mg, mbln, shTarget, 0);

    if (L == 0) {
      // bank / industry updates feed layer 2; layer-2 versions are dead code
      run_rga(pBS, 4, sh, NS, bhv, NB, eiBS, eaBS, E_BS, bhv);
      run_rga(pIS, 4, sh, NS, ihv, NI, eiIS, eaIS, E_IS, ihv);
      run_rga(pII, 4, ihv, NI, ihv, NI, eiII, eaII, E_II, ihv);
    } else {
      (void)pBS; (void)pIS; (void)pII;
    }
  }
}